// SpatialEncoding_31945966748032
// MI455X (gfx1250) — compile-verified
//
#include <hip/hip_runtime.h>

#define NPTS 4096
#define DIM  256
#define NH   8
#define TJ   4                                  // j-tiles per wave
#define TILES_PER_DIM (NPTS / 16)               // 256
#define TASKS (TILES_PER_DIM * (TILES_PER_DIM / TJ))  // 16384 wave-tasks
#define BLOCKS (TASKS / 8)                      // 8 waves per 256-thread block

typedef __attribute__((ext_vector_type(2))) float v2f;
typedef __attribute__((ext_vector_type(8))) float v8f;

// ---- monotonic float<->uint encoding so unsigned atomicMin/Max == float order
__device__ __forceinline__ unsigned fenc(float f) {
    unsigned u = __float_as_uint(f);
    return (u & 0x80000000u) ? ~u : (u | 0x80000000u);
}
__device__ __forceinline__ float fdec(unsigned k) {
    unsigned u = (k & 0x80000000u) ? (k & 0x7FFFFFFFu) : ~k;
    return __uint_as_float(u);
}

// ---- ws layout: [0..7] two uints (min,max encoded); floats at 64B offsets
#define WS_NORMS_OFF   64
#define WS_ROW_OFF     (64 + NPTS * 4)
#define WS_COL_OFF     (64 + 2 * NPTS * 4)

__global__ void init_minmax_kernel(unsigned* mm) {
    if (threadIdx.x == 0) {
        mm[0] = 0xFFFFFFFFu;  // encoded-min sentinel (max key)
        mm[1] = 0x00000000u;  // encoded-max sentinel
    }
}

__global__ __launch_bounds__(256)
void norms_kernel(const float* __restrict__ coords, float* __restrict__ norms) {
    int i = blockIdx.x * blockDim.x + threadIdx.x;
    if (i < NPTS) {
        float x = coords[3 * i + 0], y = coords[3 * i + 1], z = coords[3 * i + 2];
        norms[i] = x * x + y * y + z * z;
    }
}

__global__ __launch_bounds__(256)
void rowcol_kernel(const float* __restrict__ x, const float* __restrict__ wdist,
                   float* __restrict__ row_term, float* __restrict__ col_term) {
    __shared__ float slo[256];
    __shared__ float shi[256];
    int row = blockIdx.x, t = threadIdx.x;
    float v = x[row * DIM + t];
    slo[t] = v * wdist[t];            // weights_dist[:D]
    shi[t] = v * wdist[DIM + 1 + t];  // weights_dist[D+1:]
    __syncthreads();
    for (int s = 128; s > 0; s >>= 1) {
        if (t < s) { slo[t] += slo[t + s]; shi[t] += shi[t + s]; }
        __syncthreads();
    }
    if (t == 0) { row_term[row] = slo[0]; col_term[row] = shi[0]; }
}

// Branchless A/B fragment build (16x4 f32, K padded to 4):
// lane holds row/col = lane%16; lanes 0-15 -> K={x,y}, lanes 16-31 -> K={z,0}.
__device__ __forceinline__ v2f make_frag(float cx, float cy, float cz,
                                         bool lo, float scale) {
    v2f f;
    f.x = (lo ? cx : cz) * scale;
    f.y = lo ? (cy * scale) : 0.0f;
    return f;
}

__global__ __launch_bounds__(256)
void minmax_kernel(const float* __restrict__ coords, const float* __restrict__ wdist,
                   const float* __restrict__ norms, const float* __restrict__ row_term,
                   const float* __restrict__ col_term, unsigned* __restrict__ mm) {
    int wave = threadIdx.x >> 5;
    int lane = threadIdx.x & 31;
    int lane15 = lane & 15;
    bool lo = lane < 16;
    int task = blockIdx.x * 8 + wave;
    int ti  = (task >> 6) * 16;          // row-tile base
    int tj0 = (task & 63) * (16 * TJ);   // first col-tile base
    float w_d = wdist[DIM];

    const float* pa = coords + (ti + lane15) * 3;
    v2f a = make_frag(pa[0], pa[1], pa[2], lo, -2.0f);
    int half8 = (lane >> 4) * 8;

    const float* prn = norms + ti + half8;
    const float* prt = row_term + ti + half8;
    float rn[8], rt[8];
#pragma unroll
    for (int r = 0; r < 8; ++r) { rn[r] = prn[r]; rt[r] = prt[r]; }

    const float* pb = coords + (tj0 + lane15) * 3;   // +48 floats per j-tile
    const float* pn = norms + tj0 + lane15;          // +16 floats per j-tile
    const float* pc = col_term + tj0 + lane15;

    float lmin = 3.4e38f, lmax = -3.4e38f;
#pragma unroll
    for (int jj = 0; jj < TJ; ++jj) {
        v2f b = make_frag(pb[jj * 48 + 0], pb[jj * 48 + 1], pb[jj * 48 + 2],
                          lo, 1.0f);
        v8f c = {};
        c = __builtin_amdgcn_wmma_f32_16x16x4_f32(false, a, false, b, (short)0, c,
                                                  false, false);
        float nn = pn[jj * 16], ct = pc[jj * 16];
#pragma unroll
        for (int r = 0; r < 8; ++r) {
            float d2 = c[r] + rn[r] + nn;
            float dist = __builtin_amdgcn_sqrtf(fmaxf(d2, 0.0f));
            float s = fmaf(dist, w_d, rt[r] + ct);
            lmin = fminf(lmin, s);
            lmax = fmaxf(lmax, s);
        }
    }

    __shared__ float smn[256];
    __shared__ float smx[256];
    int t = threadIdx.x;
    smn[t] = lmin; smx[t] = lmax;
    __syncthreads();
    for (int s = 128; s > 0; s >>= 1) {
        if (t < s) {
            smn[t] = fminf(smn[t], smn[t + s]);
            smx[t] = fmaxf(smx[t], smx[t + s]);
        }
        __syncthreads();
    }
    if (t == 0) {
        atomicMin(&mm[0], fenc(smn[0]));
        atomicMax(&mm[1], fenc(smx[0]));
    }
}

__global__ __launch_bounds__(256)
void encode_kernel(const float* __restrict__ coords, const float* __restrict__ wdist,
                   const float* __restrict__ means, const float* __restrict__ stds,
                   const float* __restrict__ norms, const float* __restrict__ row_term,
                   const float* __restrict__ col_term, const unsigned* __restrict__ mm,
                   float* __restrict__ out) {
    int wave = threadIdx.x >> 5;
    int lane = threadIdx.x & 31;
    int lane15 = lane & 15;
    bool lo = lane < 16;
    int task = blockIdx.x * 8 + wave;
    int ti  = (task >> 6) * 16;
    int tj0 = (task & 63) * (16 * TJ);

    float smin = fdec(mm[0]);
    float smax = fdec(mm[1]);
    float inv  = 1.0f / (smax - smin);
    float nsc  = -smin * inv;            // t = s*inv + nsc
    float w_d  = wdist[DIM];

    // Gaussian h: c_h * 2^(-(b_h*t - m_h*b_h)^2)
    //   b_h = sqrt(0.5*log2(e))/s_h ; c_h = (1/sqrt(2pi))/(s_h*H)
    float bh[NH], mb[NH], ch[NH];
#pragma unroll
    for (int h = 0; h < NH; ++h) {
        float sd = stds[h];
        float b  = 0.84932180028801904f / sd;   // sqrt(0.5 * 1.4426950408889634)
        bh[h] = b;
        mb[h] = means[h] * b;
        ch[h] = 0.3989422804014327f / (sd * (float)NH);
    }

    const float* pa = coords + (ti + lane15) * 3;
    v2f a = make_frag(pa[0], pa[1], pa[2], lo, -2.0f);
    int half8 = (lane >> 4) * 8;

    const float* prn = norms + ti + half8;
    const float* prt = row_term + ti + half8;
    float rn[8], rt[8];
#pragma unroll
    for (int r = 0; r < 8; ++r) { rn[r] = prn[r]; rt[r] = prt[r]; }

    const float* pb = coords + (tj0 + lane15) * 3;
    const float* pn = norms + tj0 + lane15;
    const float* pc = col_term + tj0 + lane15;
    float* po = out + (size_t)(ti + half8) * NPTS + tj0 + lane15;

#pragma unroll
    for (int jj = 0; jj < TJ; ++jj) {
        v2f b = make_frag(pb[jj * 48 + 0], pb[jj * 48 + 1], pb[jj * 48 + 2],
                          lo, 1.0f);
        v8f c = {};
        c = __builtin_amdgcn_wmma_f32_16x16x4_f32(false, a, false, b, (short)0, c,
                                                  false, false);
        float nn = pn[jj * 16], ct = pc[jj * 16];
#pragma unroll
        for (int r = 0; r < 8; ++r) {
            float d2 = c[r] + rn[r] + nn;
            float dist = __builtin_amdgcn_sqrtf(fmaxf(d2, 0.0f));
            float s = fmaf(dist, w_d, rt[r] + ct);
            float t = fmaf(s, inv, nsc);
            float acc = 0.0f;
#pragma unroll
            for (int h = 0; h < NH; ++h) {
                float u = fmaf(t, bh[h], -mb[h]);
                float e = -u * u;                       // neg via src modifier
                acc = fmaf(ch[h], __builtin_amdgcn_exp2f(e), acc);
            }
            po[r * NPTS + jj * 16] = acc;               // immediate offsets
        }
    }
}

extern "C" void kernel_launch(void* const* d_in, const int* in_sizes, int n_in,
                              void* d_out, int out_size, void* d_ws, size_t ws_size,
                              hipStream_t stream) {
    const float* x      = (const float*)d_in[0];  // (4096,256)
    const float* coords = (const float*)d_in[1];  // (4096,3)
    const float* means  = (const float*)d_in[2];  // (8,)
    const float* stds   = (const float*)d_in[3];  // (8,)
    const float* wdist  = (const float*)d_in[4];  // (513,)
    float* out = (float*)d_out;

    char* ws = (char*)d_ws;
    unsigned* mm     = (unsigned*)ws;
    float* norms     = (float*)(ws + WS_NORMS_OFF);
    float* row_term  = (float*)(ws + WS_ROW_OFF);
    float* col_term  = (float*)(ws + WS_COL_OFF);

    init_minmax_kernel<<<1, 32, 0, stream>>>(mm);
    norms_kernel<<<NPTS / 256, 256, 0, stream>>>(coords, norms);
    rowcol_kernel<<<NPTS, 256, 0, stream>>>(x, wdist, row_term, col_term);
    minmax_kernel<<<BLOCKS, 256, 0, stream>>>(coords, wdist, norms,
                                              row_term, col_term, mm);
    encode_kernel<<<BLOCKS, 256, 0, stream>>>(coords, wdist, means, stds,
                                              norms, row_term, col_term, mm, out);
    (void)in_sizes; (void)n_in; (void)out_size; (void)ws_size;
}